// GatedGAT_9663676416811
// MI455X (gfx1250) — compile-verified
//
#include <hip/hip_runtime.h>
#include <hip/hip_bf16.h>
#include <math.h>

typedef __attribute__((ext_vector_type(2))) float v2f;
typedef __attribute__((ext_vector_type(8))) float v8f;

#define IN_F   256
#define MAP_F  64
#define H_F    4
#define O_F    64
#define LEAKY  0.2f

// ---------------------------------------------------------------- utilities

__global__ void fill_kernel(float* __restrict__ p, float v, size_t n) {
    size_t i = (size_t)blockIdx.x * blockDim.x + threadIdx.x;
    size_t stride = (size_t)gridDim.x * blockDim.x;
    for (; i < n; i += stride) p[i] = v;
}

__device__ __forceinline__ void atomicMaxFloat(float* addr, float val) {
    // works for accumulators initialized to -inf
    if (val >= 0.0f)
        atomicMax((int*)addr, __float_as_int(val));
    else
        atomicMin((unsigned int*)addr, __float_as_uint(val));
}

__device__ __forceinline__ float leaky(float e) {
    return e >= 0.0f ? e : LEAKY * e;
}

// ---------------------------------------------------------------- WMMA GEMM
// C[M,Nc] = A[M,K] @ B[K,Nc] (+ bias[Nc]); row-major; one wave per 16x16 tile.
// fp32 matrix cores: V_WMMA_F32_16X16X4_F32.
// A 16x4 layout: lanes 0-15 -> K={0,1}, lanes 16-31 -> K={2,3}; M = lane&15.
// B 4x16 layout: mirror (N = lane&15, K pair selected by lane>>4).
// C/D: VGPR r: lanes 0-15 -> M=r, lanes 16-31 -> M=r+8, N = lane&15.

__global__ void wmma_gemm_f32(const float* __restrict__ A,
                              const float* __restrict__ B,
                              const float* __restrict__ bias,
                              float* __restrict__ C,
                              int M, int K, int Nc) {
    int wave = blockIdx.x * (blockDim.x >> 5) + (threadIdx.x >> 5);
    int lane = threadIdx.x & 31;
    int ntn = Nc >> 4;
    int ntiles = (M >> 4) * ntn;
    if (wave >= ntiles) return;
    int tm = wave / ntn, tn = wave % ntn;
    int m  = lane & 15;
    int kg = (lane >> 4) << 1;   // 0 or 2

    const float* arow = A + (size_t)(tm * 16 + m) * K;
    const float* bcol = B + tn * 16 + m;

    v8f acc = {};
    for (int k = 0; k < K; k += 4) {
        v2f a, b;
        a.x = arow[k + kg];
        a.y = arow[k + kg + 1];
        b.x = bcol[(size_t)(k + kg) * Nc];
        b.y = bcol[(size_t)(k + kg + 1) * Nc];
        acc = __builtin_amdgcn_wmma_f32_16x16x4_f32(
            false, a, false, b, (short)0, acc, false, false);
    }

    int col  = tn * 16 + m;
    int row0 = tm * 16 + (lane >> 4) * 8;
    float bv = bias ? bias[col] : 0.0f;
#pragma unroll
    for (int r = 0; r < 8; ++r)
        C[(size_t)(row0 + r) * Nc + col] = acc[r] + bv;
}

// merge = concat(x[., 256], gated[., 64]) @ W[320,64] + b ; K = 320, Nc = 64
__global__ void wmma_merge_gemm(const float* __restrict__ x,
                                const float* __restrict__ gated,
                                const float* __restrict__ W,
                                const float* __restrict__ bias,
                                float* __restrict__ out, int M) {
    const int K = IN_F + O_F;   // 320
    const int Nc = O_F;         // 64
    int wave = blockIdx.x * (blockDim.x >> 5) + (threadIdx.x >> 5);
    int lane = threadIdx.x & 31;
    int ntn = Nc >> 4;          // 4
    int ntiles = (M >> 4) * ntn;
    if (wave >= ntiles) return;
    int tm = wave / ntn, tn = wave % ntn;
    int m  = lane & 15;
    int kg = (lane >> 4) << 1;
    int row = tm * 16 + m;
    const float* xrow = x + (size_t)row * IN_F;
    const float* grow = gated + (size_t)row * O_F;
    const float* bcol = W + tn * 16 + m;

    v8f acc = {};
    for (int k = 0; k < K; k += 4) {
        int ka = k + kg;
        v2f a, b;
        a.x = (ka     < IN_F) ? xrow[ka]     : grow[ka - IN_F];
        a.y = (ka + 1 < IN_F) ? xrow[ka + 1] : grow[ka + 1 - IN_F];
        b.x = bcol[(size_t)ka * Nc];
        b.y = bcol[(size_t)(ka + 1) * Nc];
        acc = __builtin_amdgcn_wmma_f32_16x16x4_f32(
            false, a, false, b, (short)0, acc, false, false);
    }

    int col  = tn * 16 + m;
    int row0 = tm * 16 + (lane >> 4) * 8;
    float bv = bias[col];
#pragma unroll
    for (int r = 0; r < 8; ++r)
        out[(size_t)(row0 + r) * Nc + col] = acc[r] + bv;
}

// ---------------------------------------------------------------- node kernels

__global__ void el_er_kernel(const float* __restrict__ feat,
                             const float* __restrict__ attn_l,
                             const float* __restrict__ attn_r,
                             float* __restrict__ el, float* __restrict__ er, int N) {
    int t = blockIdx.x * blockDim.x + threadIdx.x;
    if (t >= N * H_F) return;
    int n = t >> 2, h = t & 3;
    const float* f  = feat + (size_t)n * (H_F * O_F) + h * O_F;
    const float* al = attn_l + h * O_F;
    const float* ar = attn_r + h * O_F;
    float sl = 0.0f, sr = 0.0f;
#pragma unroll 8
    for (int o = 0; o < O_F; ++o) { float v = f[o]; sl += v * al[o]; sr += v * ar[o]; }
    el[t] = sl; er[t] = sr;
}

// gate = sigmoid(nft @ gate_fn_W + b); gated = mean_h gate[h]*attn_out[n,h,:]
__global__ void node_gate_kernel(const float* __restrict__ x,
                                 const float* __restrict__ sum_x,
                                 const float* __restrict__ max_z,
                                 const float* __restrict__ deg,
                                 const float* __restrict__ gfw,  // [576,4]
                                 const float* __restrict__ gfb,  // [4]
                                 const float* __restrict__ attn_out,
                                 float* __restrict__ gated, int N) {
    int n = blockIdx.x * (blockDim.x >> 5) + (threadIdx.x >> 5);
    int lane = threadIdx.x & 31;
    if (n >= N) return;
    float dg  = deg[n];
    float inv = 1.0f / fmaxf(dg, 1.0f);
    float p0 = 0.f, p1 = 0.f, p2 = 0.f, p3 = 0.f;
#pragma unroll
    for (int j = 0; j < 18; ++j) {           // 576 = 32*18
        int idx = lane + 32 * j;
        float v;
        if (idx < IN_F) {
            v = x[(size_t)n * IN_F + idx];
        } else if (idx < IN_F + MAP_F) {
            float mz = max_z[(size_t)n * MAP_F + (idx - IN_F)];
            v = (dg > 0.0f) ? mz : 0.0f;     // empty neighborhood -> 0
        } else {
            v = sum_x[(size_t)n * IN_F + (idx - IN_F - MAP_F)] * inv;  // mean_z
        }
        const float* w = gfw + (size_t)idx * H_F;
        p0 += v * w[0]; p1 += v * w[1]; p2 += v * w[2]; p3 += v * w[3];
    }
#pragma unroll
    for (int off = 16; off >= 1; off >>= 1) {
        p0 += __shfl_xor(p0, off, 32);
        p1 += __shfl_xor(p1, off, 32);
        p2 += __shfl_xor(p2, off, 32);
        p3 += __shfl_xor(p3, off, 32);
    }
    float g0 = 1.0f / (1.0f + expf(-(p0 + gfb[0])));
    float g1 = 1.0f / (1.0f + expf(-(p1 + gfb[1])));
    float g2 = 1.0f / (1.0f + expf(-(p2 + gfb[2])));
    float g3 = 1.0f / (1.0f + expf(-(p3 + gfb[3])));
    const float* ao = attn_out + (size_t)n * (H_F * O_F);
#pragma unroll
    for (int k = 0; k < 2; ++k) {
        int o = lane + 32 * k;
        float v = 0.25f * (g0 * ao[o] + g1 * ao[O_F + o] +
                           g2 * ao[2 * O_F + o] + g3 * ao[3 * O_F + o]);
        gated[(size_t)n * O_F + o] = v;
    }
}

// ---------------------------------------------------------------- edge kernels

// pass A: deg, sum_x (segment_sum of x[src]), max_z (segment_max of z[src])
__global__ void edge_gather_kernel(const int* __restrict__ src, const int* __restrict__ dst,
                                   const float* __restrict__ x, const float* __restrict__ z,
                                   float* __restrict__ deg, float* __restrict__ sum_x,
                                   float* __restrict__ max_z, int E) {
    int e = blockIdx.x * (blockDim.x >> 5) + (threadIdx.x >> 5);
    int lane = threadIdx.x & 31;
    if (e >= E) return;
    int s = src[e], d = dst[e];
    if (lane == 0) atomicAdd(deg + d, 1.0f);
    const float* xs = x + (size_t)s * IN_F;
    float* sd = sum_x + (size_t)d * IN_F;
#pragma unroll
    for (int k = 0; k < 8; ++k) {
        int i = lane + 32 * k;
        atomicAdd(sd + i, xs[i]);
    }
    const float* zs = z + (size_t)s * MAP_F;
    float* md = max_z + (size_t)d * MAP_F;
#pragma unroll
    for (int k = 0; k < 2; ++k) {
        int i = lane + 32 * k;
        atomicMaxFloat(md + i, zs[i]);
    }
}

// pass B: m = segment_max(leaky(el[src]+er[dst]))
__global__ void edge_max_kernel(const int* __restrict__ src, const int* __restrict__ dst,
                                const float* __restrict__ el, const float* __restrict__ er,
                                float* __restrict__ m, int E) {
    int t = blockIdx.x * blockDim.x + threadIdx.x;
    if (t >= E) return;
    int s = src[t], d = dst[t];
#pragma unroll
    for (int h = 0; h < H_F; ++h) {
        float e = leaky(el[s * H_F + h] + er[d * H_F + h]);
        atomicMaxFloat(m + d * H_F + h, e);
    }
}

// pass C: s = segment_sum(exp(e - m[dst]))
__global__ void edge_sum_kernel(const int* __restrict__ src, const int* __restrict__ dst,
                                const float* __restrict__ el, const float* __restrict__ er,
                                const float* __restrict__ m, float* __restrict__ ssum, int E) {
    int t = blockIdx.x * blockDim.x + threadIdx.x;
    if (t >= E) return;
    int s = src[t], d = dst[t];
#pragma unroll
    for (int h = 0; h < H_F; ++h) {
        float e  = leaky(el[s * H_F + h] + er[d * H_F + h]);
        float ex = expf(e - m[d * H_F + h]);
        atomicAdd(ssum + d * H_F + h, ex);
    }
}

// pass D: attn_out[dst] += feat[src] * alpha
__global__ void edge_scatter_kernel(const int* __restrict__ src, const int* __restrict__ dst,
                                    const float* __restrict__ el, const float* __restrict__ er,
                                    const float* __restrict__ m, const float* __restrict__ ssum,
                                    const float* __restrict__ feat,
                                    float* __restrict__ attn_out, int E) {
    int e = blockIdx.x * (blockDim.x >> 5) + (threadIdx.x >> 5);
    int lane = threadIdx.x & 31;
    if (e >= E) return;
    int s = src[e], d = dst[e];
    float alpha[H_F];
#pragma unroll
    for (int h = 0; h < H_F; ++h) {
        float ev = leaky(el[s * H_F + h] + er[d * H_F + h]);
        float ex = expf(ev - m[d * H_F + h]);
        alpha[h] = ex / fmaxf(ssum[d * H_F + h], 1e-16f);
    }
    const float* fs = feat + (size_t)s * (H_F * O_F);
    float* ad = attn_out + (size_t)d * (H_F * O_F);
#pragma unroll
    for (int k = 0; k < 8; ++k) {
        int idx = lane + 32 * k;               // idx = h*64 + o
        atomicAdd(ad + idx, fs[idx] * alpha[idx >> 6]);
    }
}

// ---------------------------------------------------------------- launch

extern "C" void kernel_launch(void* const* d_in, const int* in_sizes, int n_in,
                              void* d_out, int out_size, void* d_ws, size_t ws_size,
                              hipStream_t stream) {
    const float* x         = (const float*)d_in[0];
    const int*   src       = (const int*)  d_in[1];
    const int*   dst       = (const int*)  d_in[2];
    const float* W_gat     = (const float*)d_in[3];
    const float* attn_l    = (const float*)d_in[4];
    const float* attn_r    = (const float*)d_in[5];
    const float* gate_m_W  = (const float*)d_in[6];
    const float* gate_m_b  = (const float*)d_in[7];
    const float* gate_fn_W = (const float*)d_in[8];
    const float* gate_fn_b = (const float*)d_in[9];
    const float* merge_W   = (const float*)d_in[10];
    const float* merge_b   = (const float*)d_in[11];
    float* out = (float*)d_out;

    int N = in_sizes[0] / IN_F;
    int E = in_sizes[1];

    // workspace layout (floats)
    float* ws = (float*)d_ws;
    size_t off = 0;
    float* deg      = ws + off; off += (size_t)N;
    float* sum_x    = ws + off; off += (size_t)N * IN_F;
    float* z        = ws + off; off += (size_t)N * MAP_F;
    float* max_z    = ws + off; off += (size_t)N * MAP_F;
    float* feat     = ws + off; off += (size_t)N * H_F * O_F;
    float* el       = ws + off; off += (size_t)N * H_F;
    float* er       = ws + off; off += (size_t)N * H_F;
    float* mbuf     = ws + off; off += (size_t)N * H_F;
    float* ssum     = ws + off; off += (size_t)N * H_F;
    float* attn_out = ws + off; off += (size_t)N * H_F * O_F;
    float* gated    = ws + off; off += (size_t)N * O_F;

    const int TB = 256;
    const float NEG_INF = -INFINITY;

    // init accumulators
    fill_kernel<<<2048, TB, 0, stream>>>(deg,      0.0f,   (size_t)N);
    fill_kernel<<<2048, TB, 0, stream>>>(sum_x,    0.0f,   (size_t)N * IN_F);
    fill_kernel<<<2048, TB, 0, stream>>>(max_z,    NEG_INF,(size_t)N * MAP_F);
    fill_kernel<<<2048, TB, 0, stream>>>(mbuf,     NEG_INF,(size_t)N * H_F);
    fill_kernel<<<2048, TB, 0, stream>>>(ssum,     0.0f,   (size_t)N * H_F);
    fill_kernel<<<2048, TB, 0, stream>>>(attn_out, 0.0f,   (size_t)N * H_F * O_F);

    int wavesPerBlock = TB / 32;

    // z = x @ gate_m_W + b   [N,256]x[256,64]
    {
        int tiles = (N / 16) * (MAP_F / 16);
        int blocks = (tiles + wavesPerBlock - 1) / wavesPerBlock;
        wmma_gemm_f32<<<blocks, TB, 0, stream>>>(x, gate_m_W, gate_m_b, z, N, IN_F, MAP_F);
    }
    // feat = x @ W_gat       [N,256]x[256,256]
    {
        int tiles = (N / 16) * ((H_F * O_F) / 16);
        int blocks = (tiles + wavesPerBlock - 1) / wavesPerBlock;
        wmma_gemm_f32<<<blocks, TB, 0, stream>>>(x, W_gat, nullptr, feat, N, IN_F, H_F * O_F);
    }
    // el, er
    el_er_kernel<<<(N * H_F + TB - 1) / TB, TB, 0, stream>>>(feat, attn_l, attn_r, el, er, N);

    // edge passes
    int edgeWaveBlocks = (E + wavesPerBlock - 1) / wavesPerBlock;
    edge_gather_kernel<<<edgeWaveBlocks, TB, 0, stream>>>(src, dst, x, z, deg, sum_x, max_z, E);
    edge_max_kernel<<<(E + TB - 1) / TB, TB, 0, stream>>>(src, dst, el, er, mbuf, E);
    edge_sum_kernel<<<(E + TB - 1) / TB, TB, 0, stream>>>(src, dst, el, er, mbuf, ssum, E);
    edge_scatter_kernel<<<edgeWaveBlocks, TB, 0, stream>>>(src, dst, el, er, mbuf, ssum,
                                                           feat, attn_out, E);

    // gate + gated merge input
    node_gate_kernel<<<(N + wavesPerBlock - 1) / wavesPerBlock, TB, 0, stream>>>(
        x, sum_x, max_z, deg, gate_fn_W, gate_fn_b, attn_out, gated, N);

    // out = concat(x, gated) @ merge_W + merge_b
    {
        int tiles = (N / 16) * (O_F / 16);
        int blocks = (tiles + wavesPerBlock - 1) / wavesPerBlock;
        wmma_merge_gemm<<<blocks, TB, 0, stream>>>(x, gated, merge_W, merge_b, out, N);
    }
}